// HeytingNeuralLayer_77945066488172
// MI455X (gfx1250) — compile-verified
//
#include <hip/hip_runtime.h>

// Heyting forward: out[b,n] = min_k ( s(x[b,k]) <= s(w[n,k]) ? 1.0f : s(w[n,k]) )
// where s() = sigmoid. (stop_gradient(exact-soft)+soft == exact in forward.)

#define BATCH  128
#define IN_F   1024
#define OUT_F  1024

#define BT       16               // batch rows per block
#define NT       8                // output cols per block (one per wave)
#define NTHREADS 256              // 8 waves of 32 (wave32)
#define KF4      (IN_F / 4)       // 256 float4 per row
#define XF4      (BT * KF4)       // 4096 float4  (64 KB)
#define WF4      (NT * KF4)       // 2048 float4  (32 KB)
#define XITER    (XF4 / NTHREADS) // 16 async b128 per thread for x tile
#define WITER    (WF4 / NTHREADS) // 8 async b128 per thread for w tile
#define KSWEEPS  (KF4 / 32)       // 8 lane-sweeps over one row

typedef float f4  __attribute__((ext_vector_type(4)));
typedef int   v4i __attribute__((ext_vector_type(4)));

// --- CDNA5 async global->LDS path (gfx1250), with portable fallback ---------
#if defined(__AMDGCN__) && defined(__gfx1250__) && \
    __has_builtin(__builtin_amdgcn_global_load_async_to_lds_b128) && \
    __has_builtin(__builtin_amdgcn_s_wait_asynccnt)
#define HAVE_ASYNC_LDS 1
// builtin signature (from clang diagnostics): (as1 v4i*, as3 v4i*, imm offset, imm cpol)
typedef __attribute__((address_space(1))) v4i* gptr_v4i;
typedef __attribute__((address_space(3))) v4i* lptr_v4i;

// Compile-time unrolled async copy: the IOFFSET immediate is added to BOTH the
// global and the LDS address (ISA 10.x async pseudocode), so one base pair +
// constant offsets performs a flat lockstep copy of N*16 bytes per thread (N*STEP per block).
template <int N, int STEP>
struct AsyncCopy {
    static __device__ __forceinline__ void run(gptr_v4i g, lptr_v4i l) {
        AsyncCopy<N - 1, STEP>::run(g, l);
        __builtin_amdgcn_global_load_async_to_lds_b128(g, l, (N - 1) * STEP, 0);
    }
};
template <int STEP>
struct AsyncCopy<0, STEP> {
    static __device__ __forceinline__ void run(gptr_v4i, lptr_v4i) {}
};
#endif

__device__ __forceinline__ float sigmoid_f(float v) {
    return 1.0f / (1.0f + __expf(-v));
}

__device__ __forceinline__ f4 sigmoid_f4(f4 v) {
    f4 r;
    r.x = sigmoid_f(v.x);
    r.y = sigmoid_f(v.y);
    r.z = sigmoid_f(v.z);
    r.w = sigmoid_f(v.w);
    return r;
}

__global__ __launch_bounds__(NTHREADS)
void heyting_fwd_kernel(const float* __restrict__ x,
                        const float* __restrict__ w,
                        float* __restrict__ out) {
    __shared__ f4 lx[XF4];   // 64 KB: sigmoid(x) tile, BT rows x IN_F
    __shared__ f4 lw[WF4];   // 32 KB: sigmoid(w) tile, NT rows x IN_F

    const int tid  = threadIdx.x;
    const int lane = tid & 31;
    const int wave = tid >> 5;                 // 0..7 -> output column within tile
    const int n0   = blockIdx.x * NT;
    const int b0   = blockIdx.y * BT;

    // Both tiles are fully contiguous ranges of global memory (whole rows).
    const f4* xq = reinterpret_cast<const f4*>(x) + (size_t)b0 * KF4;
    const f4* wq = reinterpret_cast<const f4*>(w) + (size_t)n0 * KF4;

    // ---- Stage raw tiles into LDS ----
#if defined(HAVE_ASYNC_LDS)
    AsyncCopy<XITER, NTHREADS * 16>::run((gptr_v4i)(xq + tid),
                                         (lptr_v4i)(&lx[tid]));
    AsyncCopy<WITER, NTHREADS * 16>::run((gptr_v4i)(wq + tid),
                                         (lptr_v4i)(&lw[tid]));
    __builtin_amdgcn_s_wait_asynccnt(0);
#else
#pragma unroll
    for (int j = 0; j < XITER; ++j) lx[tid + j * NTHREADS] = xq[tid + j * NTHREADS];
#pragma unroll
    for (int j = 0; j < WITER; ++j) lw[tid + j * NTHREADS] = wq[tid + j * NTHREADS];
#endif
    __syncthreads();

    // ---- In-place sigmoid transform (once per staged element) ----
#pragma unroll
    for (int j = 0; j < XITER; ++j) {
        const int i = tid + j * NTHREADS;
        lx[i] = sigmoid_f4(lx[i]);
    }
#pragma unroll
    for (int j = 0; j < WITER; ++j) {
        const int i = tid + j * NTHREADS;
        lw[i] = sigmoid_f4(lw[i]);
    }
    __syncthreads();

    // ---- Compute: wave `wave` owns output column n = n0 + wave ----
    const int n = n0 + wave;

    // Preload this wave's weight row: 8 x float4 per lane covers all K=1024.
    f4 wr[KSWEEPS];
#pragma unroll
    for (int s = 0; s < KSWEEPS; ++s)
        wr[s] = lw[wave * KF4 + s * 32 + lane];

    for (int b = 0; b < BT; ++b) {
        f4 acc;
        acc.x = 1.0f; acc.y = 1.0f; acc.z = 1.0f; acc.w = 1.0f;
#pragma unroll
        for (int s = 0; s < KSWEEPS; ++s) {
            // conflict-free: consecutive lanes read consecutive 16B of one row
            f4 xv = lx[b * KF4 + s * 32 + lane];
            f4 wv = wr[s];
            // impl = (xl <= wl) ? 1 : wl ; acc = min(acc, impl); acc<=1 so:
            acc.x = (xv.x <= wv.x) ? acc.x : fminf(acc.x, wv.x);
            acc.y = (xv.y <= wv.y) ? acc.y : fminf(acc.y, wv.y);
            acc.z = (xv.z <= wv.z) ? acc.z : fminf(acc.z, wv.z);
            acc.w = (xv.w <= wv.w) ? acc.w : fminf(acc.w, wv.w);
        }
        float m = fminf(fminf(acc.x, acc.y), fminf(acc.z, acc.w));
        // cross-lane min over the 32 lanes (each lane held a k-subset)
#pragma unroll
        for (int off = 16; off > 0; off >>= 1)
            m = fminf(m, __shfl_xor(m, off, 32));
        if (lane == 0)
            out[(size_t)(b0 + b) * OUT_F + n] = m;
    }
}

extern "C" void kernel_launch(void* const* d_in, const int* in_sizes, int n_in,
                              void* d_out, int out_size, void* d_ws, size_t ws_size,
                              hipStream_t stream) {
    (void)in_sizes; (void)n_in; (void)d_ws; (void)ws_size; (void)out_size;
    const float* x = (const float*)d_in[0];   // (128, 1024) f32
    const float* w = (const float*)d_in[1];   // (1024, 1024) f32
    float* out = (float*)d_out;               // (128, 1024) f32

    dim3 grid(OUT_F / NT, BATCH / BT);        // (128, 8)
    dim3 block(NTHREADS);
    heyting_fwd_kernel<<<grid, block, 0, stream>>>(x, w, out);
}